// SampleModel_57415122813019
// MI455X (gfx1250) — compile-verified
//
#include <hip/hip_runtime.h>

// GCN layer for MI455X (gfx1250, wave32):
//   out = segment_sum( (x@W)[src] * coef ) + bias,  coef = rsqrt(deg_d)*rsqrt(deg_s)
// Project first (linear aggregation) -> per-edge payload drops 128->64 floats.
// GEMM uses V_WMMA_F32_16X16X4_F32 (full-precision f32 matrix core path).

#define N_NODES 50000
#define N_EDGES 600000
#define IN_CH   128
#define OUT_CH  64
#define N_TILES (N_NODES / 16)   // 3125, exact

typedef __attribute__((ext_vector_type(2))) float v2f;
typedef __attribute__((ext_vector_type(8))) float v8f;

// ---- degree passes -------------------------------------------------------
__global__ void init_deg_kernel(float* __restrict__ deg) {
    int i = blockIdx.x * blockDim.x + threadIdx.x;
    if (i < N_NODES) deg[i] = 1.0f;          // self loop
}

__global__ void count_deg_kernel(const int* __restrict__ dst, float* __restrict__ deg) {
    int e = blockIdx.x * blockDim.x + threadIdx.x;
    if (e < N_EDGES) atomicAdd(&deg[dst[e]], 1.0f);
}

__global__ void rsqrt_deg_kernel(float* __restrict__ deg) {
    int i = blockIdx.x * blockDim.x + threadIdx.x;
    if (i < N_NODES) deg[i] = rsqrtf(deg[i]);   // deg[] becomes isd[]
}

// ---- y = x @ W via V_WMMA_F32_16X16X4_F32 --------------------------------
// One wave per 16-row tile; 4 accumulators cover all 64 output columns.
// A 16x4 f32 layout: lane l holds A[l&15][2*(l>>4) + {0,1}] in 2 VGPRs.
// B 4x16  f32 layout: lane l holds B[2*(l>>4) + {0,1}][l&15].
// C/D 16x16: vgpr v, lane l -> C[v + 8*(l>>4)][l&15].
__global__ __launch_bounds__(128) void gemm_wmma_kernel(
    const float* __restrict__ x, const float* __restrict__ w, float* __restrict__ y) {
    const int wave = threadIdx.x >> 5;
    const int lane = threadIdx.x & 31;
    const int tile = blockIdx.x * 4 + wave;
    if (tile >= N_TILES) return;
    const int rowBase = tile * 16;
    const int m  = lane & 15;
    const int kh = lane >> 4;                // 0 or 1: which K-pair this half-wave holds

    v8f acc0 = {}, acc1 = {}, acc2 = {}, acc3 = {};
    const float* xrow = x + (size_t)(rowBase + m) * IN_CH + 2 * kh;

    for (int k0 = 0; k0 < IN_CH; k0 += 4) {
        v2f a;
        a.x = xrow[k0];
        a.y = xrow[k0 + 1];
        const float* bp = w + (size_t)(k0 + 2 * kh) * OUT_CH + m;
        v2f b0; b0.x = bp[0];  b0.y = bp[OUT_CH + 0];
        v2f b1; b1.x = bp[16]; b1.y = bp[OUT_CH + 16];
        v2f b2; b2.x = bp[32]; b2.y = bp[OUT_CH + 32];
        v2f b3; b3.x = bp[48]; b3.y = bp[OUT_CH + 48];
        acc0 = __builtin_amdgcn_wmma_f32_16x16x4_f32(false, a, false, b0, (short)0, acc0, false, false);
        acc1 = __builtin_amdgcn_wmma_f32_16x16x4_f32(false, a, false, b1, (short)0, acc1, false, false);
        acc2 = __builtin_amdgcn_wmma_f32_16x16x4_f32(false, a, false, b2, (short)0, acc2, false, false);
        acc3 = __builtin_amdgcn_wmma_f32_16x16x4_f32(false, a, false, b3, (short)0, acc3, false, false);
    }

    float* yb = y + (size_t)(rowBase + 8 * kh) * OUT_CH + m;
#pragma unroll
    for (int v = 0; v < 8; v++) {
        yb[(size_t)v * OUT_CH + 0]  = acc0[v];
        yb[(size_t)v * OUT_CH + 16] = acc1[v];
        yb[(size_t)v * OUT_CH + 32] = acc2[v];
        yb[(size_t)v * OUT_CH + 48] = acc3[v];
    }
}

// ---- out init: bias + self-loop term (coef_ii = isd[i]^2) ----------------
__global__ void selfloop_bias_kernel(const float* __restrict__ y,
                                     const float* __restrict__ isd,
                                     const float* __restrict__ bias,
                                     float* __restrict__ out) {
    int t = blockIdx.x * blockDim.x + threadIdx.x;
    if (t >= N_NODES * OUT_CH) return;
    int i = t >> 6;            // OUT_CH == 64
    int c = t & 63;
    float s = isd[i];
    out[t] = bias[c] + y[t] * s * s;
}

// ---- edge scatter-add: out[dst] += y[src] * isd[src]*isd[dst] ------------
// y (12.8 MB) and out (12.8 MB) both fit in the 192 MB L2, so the random
// gathers and f32 atomics stay on-die.
__global__ void aggregate_kernel(const int* __restrict__ src, const int* __restrict__ dst,
                                 const float* __restrict__ y, const float* __restrict__ isd,
                                 float* __restrict__ out) {
    long long t = (long long)blockIdx.x * blockDim.x + threadIdx.x;
    if (t >= (long long)N_EDGES * OUT_CH) return;
    int e = (int)(t >> 6);
    int c = (int)(t & 63);
    int s = src[e], d = dst[e];
    float coef = isd[s] * isd[d];
    atomicAdd(out + (size_t)d * OUT_CH + c, y[(size_t)s * OUT_CH + c] * coef);
}

extern "C" void kernel_launch(void* const* d_in, const int* in_sizes, int n_in,
                              void* d_out, int out_size, void* d_ws, size_t ws_size,
                              hipStream_t stream) {
    const float* x    = (const float*)d_in[0];          // [N_NODES, IN_CH]
    const int*   edge = (const int*)d_in[1];            // [2, N_EDGES]
    const float* w    = (const float*)d_in[2];          // [IN_CH, OUT_CH]
    const float* bias = (const float*)d_in[3];          // [OUT_CH]
    float*       out  = (float*)d_out;                  // [N_NODES, OUT_CH]

    const int* src = edge;
    const int* dst = edge + N_EDGES;

    // workspace layout: isd (50000 f32, padded to 50176) | y (50000*64 f32)
    float* deg = (float*)d_ws;
    float* y   = deg + 50176;

    init_deg_kernel <<<(N_NODES + 255) / 256, 256, 0, stream>>>(deg);
    count_deg_kernel<<<(N_EDGES + 255) / 256, 256, 0, stream>>>(dst, deg);
    rsqrt_deg_kernel<<<(N_NODES + 255) / 256, 256, 0, stream>>>(deg);

    gemm_wmma_kernel<<<(N_TILES + 3) / 4, 128, 0, stream>>>(x, w, y);

    selfloop_bias_kernel<<<(N_NODES * OUT_CH + 255) / 256, 256, 0, stream>>>(y, deg, bias, out);

    long long agg_threads = (long long)N_EDGES * OUT_CH;
    aggregate_kernel<<<(unsigned)((agg_threads + 255) / 256), 256, 0, stream>>>(src, dst, y, deg, out);
}